// KNN_11003706212674
// MI455X (gfx1250) — compile-verified
//
#include <hip/hip_runtime.h>
#include <math.h>

// KNN with density re-weighting, fused streaming top-k.
// B=4, N=8192, K=16, K2=32. Output: [B][N][K] int32 neighbor indices.
//
// Distance dot-products computed with V_WMMA_F32_16X16X4_F32:
//   A = 16x4 tile of query points (x,y,z,0), B = 4x16 tile of candidates.
// Streaming score s(n) = ||p_n||^2 - 2 q.p_n  (order-equivalent to ||q-p||^2
// for fixed q); exact distances recomputed for the surviving 32 candidates.

typedef __attribute__((ext_vector_type(2))) float v2f;
typedef __attribute__((ext_vector_type(4))) float v4f;
typedef __attribute__((ext_vector_type(8))) float v8f;

#define NPTS  8192
#define NBATCH 4
#define KOUT  16
#define K2    32
#define WAVES 8   // waves per block; block = 256 threads = 128 queries

__global__ __launch_bounds__(256) void knn_wmma_kernel(
    const float* __restrict__ xyz, int* __restrict__ out)
{
    // 64KB LDS, time-multiplexed:
    //  phase 1: per-wave 16x32 score tile at smem + wave*512
    //  phase 2: per-wave score lists (floats) at smem + wave*1024,
    //           per-wave index lists (ints)  at smem + 8192 + wave*1024
    __shared__ float smem[16384];

    const int tid  = threadIdx.x;
    const int wave = tid >> 5;
    const int lane = tid & 31;
    const int lm   = lane & 15;
    const bool hi  = lane >= 16;

    const int blocks_per_batch = NPTS / (WAVES * 16);   // 64
    const int batch  = blockIdx.x / blocks_per_batch;
    const int mblk   = blockIdx.x % blocks_per_batch;
    const int m_base = mblk * (WAVES * 16) + wave * 16;

    const float* __restrict__ P = xyz + (size_t)batch * NPTS * 3;

    float* tile = smem + wave * 512;   // 16 rows (queries) x 32 cols (candidates)

    // ---- A matrix: this wave's 16 queries, padded to K=4 ----
    // 32-bit 16x4 A layout: lanes 0-15 hold K=0 (v0) and K=1 (v1) of M=lane;
    // lanes 16-31 hold K=2 and K=3 of M=lane-16.
    const int   mq = m_base + lm;
    const float qx = P[mq * 3 + 0];
    const float qy = P[mq * 3 + 1];
    const float qz = P[mq * 3 + 2];
    v2f A;
    A.x = hi ? qz : qx;     // K=2 : K=0
    A.y = hi ? 0.0f : qy;   // K=3 : K=1

    // ---- per-lane running top-32 (unsorted, max-tracked, VGPR-resident) ----
    // Lane l covers query (l&15) and candidate half (l>=16).
    float bd[K2];
    int   bi[K2];
#pragma unroll
    for (int i = 0; i < K2; ++i) { bd[i] = 3.0e38f; bi[i] = 0; }
    float worstv = 3.0e38f;
    int   worstp = 0;

    for (int n0 = 0; n0 < NPTS; n0 += 32) {
        if (n0 + 256 < NPTS)   // WGP-scope temporal prefetch of future candidates
            __builtin_prefetch(P + (size_t)(n0 + 256) * 3 + lane * 3, 0, 3);

        // ---- two B tiles: candidates n0..n0+15 and n0+16..n0+31 ----
        const int nA = n0 + lm;
        const int nB = n0 + 16 + lm;
        const float ax = P[nA * 3 + 0], ay = P[nA * 3 + 1], az = P[nA * 3 + 2];
        const float bx = P[nB * 3 + 0], by = P[nB * 3 + 1], bz = P[nB * 3 + 2];
        const float sqA = ax * ax + ay * ay + az * az;
        const float sqB = bx * bx + by * by + bz * bz;
        v2f B0, B1;
        B0.x = hi ? az : ax;  B0.y = hi ? 0.0f : ay;
        B1.x = hi ? bz : bx;  B1.y = hi ? 0.0f : by;

        v8f c0 = {};
        v8f c1 = {};
        // 8 args: (neg_a, A, neg_b, B, c_mod, C, reuse_a, reuse_b)
        c0 = __builtin_amdgcn_wmma_f32_16x16x4_f32(false, A, false, B0,
                                                   (short)0, c0, false, false);
        c1 = __builtin_amdgcn_wmma_f32_16x16x4_f32(false, A, false, B1,
                                                   (short)0, c1, false, false);

        // ---- transpose C through LDS: tile[m][c] = ||p||^2 - 2 dot ----
        // C layout: VGPR v, lanes 0-15 -> (M=v, N=lane); lanes 16-31 -> (M=v+8, N=lane-16)
        const int mrow = hi ? 8 : 0;
#pragma unroll
        for (int v = 0; v < 8; ++v) {
            tile[(mrow + v) * 32 + lm]      = sqA - 2.0f * c0[v];
            tile[(mrow + v) * 32 + 16 + lm] = sqB - 2.0f * c1[v];
        }
        __asm__ volatile("" ::: "memory");  // same-wave DS ops are in-order in HW

        // ---- selection: lane l scans its 16 candidates for query (l&15) ----
        // Pull the whole row into registers with 4x ds_load_b128, then scan.
        const int cbase = hi ? 16 : 0;
        const v4f* rowv = (const v4f*)(tile + lm * 32 + cbase);  // 16B aligned
        float rv[16];
#pragma unroll
        for (int q = 0; q < 4; ++q) {
            const v4f t4 = rowv[q];
            rv[4 * q + 0] = t4.x;
            rv[4 * q + 1] = t4.y;
            rv[4 * q + 2] = t4.z;
            rv[4 * q + 3] = t4.w;
        }
#pragma unroll
        for (int j = 0; j < 16; ++j) {
            const float sc = rv[j];
            if (sc < worstv) {               // rare path (cmpx + execz skip)
                bd[worstp] = sc;
                bi[worstp] = n0 + cbase + j;
                worstv = -3.0e38f;
                for (int t = 0; t < K2; ++t)
                    if (bd[t] > worstv) { worstv = bd[t]; worstp = t; }
            }
        }
        __asm__ volatile("" ::: "memory");
    }

    __syncthreads();  // retire tile region before repurposing smem for lists

    float* lsc = smem + wave * 1024;                 // 32 lanes x 32 scores
    int*   lid = (int*)(smem + 8192) + wave * 1024;  // 32 lanes x 32 indices
#pragma unroll
    for (int i = 0; i < K2; ++i) {
        lsc[lane * K2 + i] = bd[i];
        lid[lane * K2 + i] = bi[i];
    }
    __asm__ volatile("" ::: "memory");

    if (lane < 16) {
        const int m = lm;

        // ---- merge the two 32-lists for this query, keep 32 smallest ----
        float d[2 * K2];
        int   id[2 * K2];
        for (int i = 0; i < K2; ++i) {
            d[i]       = lsc[m * K2 + i];
            id[i]      = lid[m * K2 + i];
            d[K2 + i]  = lsc[(m + 16) * K2 + i];
            id[K2 + i] = lid[(m + 16) * K2 + i];
        }
        for (int j = 0; j < K2; ++j) {          // partial selection sort (asc)
            int bp = j; float bv = d[j];
            for (int t = j + 1; t < 2 * K2; ++t)
                if (d[t] < bv) { bv = d[t]; bp = t; }
            float td = d[j]; d[j] = d[bp]; d[bp] = td;
            int   ti = id[j]; id[j] = id[bp]; id[bp] = ti;
        }

        // ---- density re-weighting over the 32 survivors ----
        float lx[K2], ly[K2], lz[K2], dd[K2];
        float sx = 0, sy = 0, sz = 0, sxx = 0, syy = 0, szz = 0;
        for (int i = 0; i < K2; ++i) {
            const int n = id[i];
            const float px = P[n * 3 + 0], py = P[n * 3 + 1], pz = P[n * 3 + 2];
            lx[i] = px; ly[i] = py; lz[i] = pz;
            const float dx = qx - px, dy = qy - py, dz = qz - pz;
            dd[i] = dx * dx + dy * dy + dz * dz;   // exact ||q-p||^2
            sx += px; sy += py; sz += pz;
            sxx += px * px; syy += py * py; szz += pz * pz;
        }
        const float inv  = 1.0f / (float)K2;
        const float mx = sx * inv, my = sy * inv, mz = sz * inv;
        const float vfac = 1.0f / (float)(K2 - 1);           // unbiased (ddof=1)
        const float vx = fmaxf((sxx - (float)K2 * mx * mx) * vfac, 0.0f);
        const float vy = fmaxf((syy - (float)K2 * my * my) * vfac, 0.0f);
        const float vz = fmaxf((szz - (float)K2 * mz * mz) * vfac, 0.0f);
        const float denom = (sqrtf(vx) + sqrtf(vy) + sqrtf(vz) + 3.0e-6f) * (1.0f / 3.0f);
        const float rden = -1.0f / denom;

        float wsc[K2];
        bool  used[K2];
        for (int i = 0; i < K2; ++i) {
            const float ex = lx[i] - mx, ey = ly[i] - my, ez = lz[i] - mz;
            wsc[i] = dd[i] * __expf((ex * ex + ey * ey + ez * ez) * rden);
            used[i] = false;
        }

        // ---- stable pick of 16 smallest weighted distances ----
        const int obase = (batch * NPTS + (m_base + m)) * KOUT;
        for (int k = 0; k < KOUT; ++k) {
            int bp = 0; float bv = 3.4e38f;
            for (int t = 0; t < K2; ++t)
                if (!used[t] && wsc[t] < bv) { bv = wsc[t]; bp = t; }
            used[bp] = true;
            out[obase + k] = id[bp];
        }
    }
}

extern "C" void kernel_launch(void* const* d_in, const int* in_sizes, int n_in,
                              void* d_out, int out_size, void* d_ws, size_t ws_size,
                              hipStream_t stream)
{
    (void)in_sizes; (void)n_in; (void)out_size; (void)d_ws; (void)ws_size;
    const float* xyz = (const float*)d_in[0];
    int* out = (int*)d_out;

    const int blocks = NBATCH * (NPTS / (WAVES * 16));  // 4 * 64 = 256
    knn_wmma_kernel<<<blocks, WAVES * 32, 0, stream>>>(xyz, out);
}